// MM4d_36086315221298
// MI455X (gfx1250) — compile-verified
//
#include <hip/hip_runtime.h>
#include <hip/hip_bf16.h>
#include <math.h>

typedef __attribute__((ext_vector_type(2))) float v2f;
typedef __attribute__((ext_vector_type(8))) float v8f;

#define L_    512
#define NIN_  32
#define NH_   64
#define NS_   128
#define DTR_  4
#define NO_   64
#define CH_   16   // scan chunk length

__device__ __forceinline__ float sigmoidf_(float x) { return 1.0f / (1.0f + __expf(-x)); }
__device__ __forceinline__ float siluf_(float x)    { return x * sigmoidf_(x); }
__device__ __forceinline__ float softplusf_(float x) {
  if (x > 20.0f) return x;
  return __logf(1.0f + __expf(x));
}

// ---- CDNA5 async global->LDS copy (bypasses VGPRs, tracked by ASYNCcnt) ----
// LDS address from the vdst VGPR, global address from 64-bit vaddr; 16B payload.
__device__ __forceinline__ void async_copy_b128(void* lds_dst, const void* gsrc) {
  unsigned l = (unsigned)(unsigned long long)(uintptr_t)lds_dst; // addr[31:0] = LDS offset
  asm volatile("global_load_async_to_lds_b128 %0, %1, off"
               :: "v"(l), "v"(gsrc) : "memory");
}
__device__ __forceinline__ void async_wait0() {
  asm volatile("s_wait_asynccnt 0x0" ::: "memory");
}

// One 16x16 f32 output tile per wave32, accumulating over K in steps of 4 via
// V_WMMA_F32_16X16X4_F32.  Per-lane operand layout (ISA 7.12.2):
//   A 16x4 :  lane l -> row m=l&15, VGPR j holds K = 2*(l>>4)+j
//   B 4x16 :  lane l -> col n=l&15, VGPR j holds K = 2*(l>>4)+j
//   C/D    :  VGPR i, lane l -> D[i + 8*(l>>4)][l&15]
__device__ __forceinline__ v8f wmma_tile(const float* A, int lda,
                                         const float* B, int ldb,
                                         int K, v8f c) {
  const int lane = threadIdx.x & 31;
  const int m  = lane & 15;
  const int ko = (lane >> 4) << 1;
  for (int k = 0; k < K; k += 4) {
    v2f av, bv;
    av.x = A[m * lda + k + ko];
    av.y = A[m * lda + k + ko + 1];
    bv.x = B[(k + ko) * ldb + m];
    bv.y = B[(k + ko + 1) * ldb + m];
    c = __builtin_amdgcn_wmma_f32_16x16x4_f32(false, av, false, bv,
                                              (short)0, c, false, false);
  }
  return c;
}

// ---------------- Kernel A: x = in@W_dim + b ; xz = x@W_in -> xi, z ----------
__global__ __launch_bounds__(256) void k_inproj(
    const float* __restrict__ gin, const float* __restrict__ Wdim,
    const float* __restrict__ bdim, const float* __restrict__ Win,
    float* __restrict__ xi_g, float* __restrict__ z_g) {
  __shared__ __align__(16) float s_in[32 * NIN_];
  __shared__ __align__(16) float s_x [32 * NH_];
  const int seq = blockIdx.x >> 4;
  const int l0  = (blockIdx.x & 15) * 32;
  const int tid = threadIdx.x;
  const int lane = tid & 31, w = tid >> 5;
  const int m = lane & 15, hh = lane >> 4;

  // async-stage the 32x32 input tile (4 KB, one B128 per thread)
  const float* src = gin + (size_t)seq * L_ * NIN_ + (size_t)l0 * NIN_;
  async_copy_b128(&s_in[tid * 4], src + tid * 4);
  async_wait0();
  __syncthreads();

  { // X tile (32x64) = in(32x32) @ W_dim(32x64) + b_dim
    int mi = w >> 2, ni = w & 3;
    v8f c = {};
    c = wmma_tile(&s_in[mi * 16 * NIN_], NIN_, Wdim + ni * 16, NH_, NIN_, c);
    float bb = bdim[ni * 16 + m];
#pragma unroll
    for (int i = 0; i < 8; i++)
      s_x[(mi * 16 + i + 8 * hh) * NH_ + ni * 16 + m] = c[i] + bb;
  }
  __syncthreads();

  // XZ (32x128) = X(32x64) @ W_in(64x128); cols [0,64)=xi, [64,128)=z
  for (int t = 0; t < 2; t++) {
    int idx = w * 2 + t;
    int mi = idx >> 3, nj = idx & 7;
    v8f c = {};
    c = wmma_tile(&s_x[mi * 16 * NH_], NH_, Win + nj * 16, 2 * NH_, NH_, c);
#pragma unroll
    for (int i = 0; i < 8; i++) {
      int row = l0 + mi * 16 + i + 8 * hh;
      int col = nj * 16 + m;
      float v = c[i];
      if (col < NH_) xi_g[(size_t)seq * L_ * NH_ + (size_t)row * NH_ + col] = v;
      else           z_g [(size_t)seq * L_ * NH_ + (size_t)row * NH_ + (col - NH_)] = v;
    }
  }
}

// -------- Kernel B: fused conv+SiLU+x_dbl(WMMA)+softplus + selective scan ----
__global__ __launch_bounds__(256) void k_scan(
    const float* __restrict__ xi_g, const float* __restrict__ z_g,
    const float* __restrict__ convw, const float* __restrict__ convb,
    const float* __restrict__ Wxp,  const float* __restrict__ Wdt,
    const float* __restrict__ bdt,  const float* __restrict__ Alog,
    const float* __restrict__ Dskip, float* __restrict__ y1_g) {
  __shared__ __align__(16) float s_xprev[NH_];
  __shared__ __align__(16) float s_xi[CH_ * NH_];
  __shared__ __align__(16) float s_u [CH_ * NH_];
  __shared__ __align__(16) float s_z [CH_ * NH_];
  __shared__ __align__(16) float s_dt[CH_ * NH_];
  __shared__ __align__(16) float s_bc[CH_ * 2 * NS_];   // cols [0,128)=B, [128,256)=C
  __shared__ __align__(16) float s_dpre[CH_ * DTR_];

  const int seq = blockIdx.x;
  const int tid = threadIdx.x;
  const int lane = tid & 31, w = tid >> 5;
  const int m = lane & 15, hh = lane >> 4;
  const int d = tid >> 2, g = tid & 3;      // thread owns (d, n in [32g,32g+32))

  float Arow[32], h[32];
#pragma unroll
  for (int i = 0; i < 32; i++) {
    Arow[i] = -__expf(Alog[d * NS_ + g * 32 + i]);
    h[i] = 0.0f;
  }
  const float Dd = Dskip[d];

  if (tid < NH_) s_xprev[tid] = 0.0f;
  const float* xi_s = xi_g + (size_t)seq * L_ * NH_;
  const float* z_s  = z_g  + (size_t)seq * L_ * NH_;
  float*       y1_s = y1_g + (size_t)seq * L_ * NH_;
  __syncthreads();

  for (int ck = 0; ck < L_ / CH_; ck++) {
    const int l0 = ck * CH_;
    // async-stage xi,z chunk directly into LDS (4 KB each, B128 per thread)
    async_copy_b128(&s_xi[tid * 4], xi_s + l0 * NH_ + tid * 4);
    async_copy_b128(&s_z [tid * 4], z_s  + l0 * NH_ + tid * 4);
    if (ck + 1 < L_ / CH_) {
      __builtin_prefetch(&xi_s[(l0 + CH_) * NH_ + tid], 0, 3);
      __builtin_prefetch(&z_s [(l0 + CH_) * NH_ + tid], 0, 3);
    }
    async_wait0();
    __syncthreads();

    // causal depthwise conv (K=2) + SiLU
    for (int i = tid; i < CH_ * NH_; i += 256) {
      int r = i >> 6, dd = i & 63;
      float xm1 = (r == 0) ? s_xprev[dd] : s_xi[(r - 1) * NH_ + dd];
      float up = convb[dd] + xm1 * convw[dd * 2 + 0] + s_xi[r * NH_ + dd] * convw[dd * 2 + 1];
      s_u[i] = siluf_(up);
    }
    __syncthreads();
    if (tid < NH_) s_xprev[tid] = s_xi[(CH_ - 1) * NH_ + tid];

    // x_dbl = u(16x64) @ W_xp(64x260): cols 4..259 via WMMA -> B,C ; cols 0..3 -> dpre
    for (int t = 0; t < 2; t++) {
      int nj = w + t * 8;
      v8f c = {};
      c = wmma_tile(s_u, NH_, Wxp + 4 + nj * 16, DTR_ + 2 * NS_, NH_, c);
#pragma unroll
      for (int i = 0; i < 8; i++)
        s_bc[(i + 8 * hh) * (2 * NS_) + nj * 16 + m] = c[i];
    }
    if (tid < CH_ * DTR_) {
      int r = tid >> 2, t = tid & 3;
      float acc = 0.0f;
      for (int k = 0; k < NH_; k++) acc += s_u[r * NH_ + k] * Wxp[k * (DTR_ + 2 * NS_) + t];
      s_dpre[tid] = acc;
    }
    __syncthreads();

    // delta = softplus(dpre @ W_dt + b_dt)
    for (int i = tid; i < CH_ * NH_; i += 256) {
      int r = i >> 6, dd = i & 63;
      float acc = bdt[dd];
#pragma unroll
      for (int t = 0; t < DTR_; t++) acc += s_dpre[r * DTR_ + t] * Wdt[t * NH_ + dd];
      s_dt[i] = softplusf_(acc);
    }
    __syncthreads();

    // sequential recurrence over the chunk; h stays in VGPRs
    for (int s = 0; s < CH_; s++) {
      float dt = s_dt[s * NH_ + d];
      float us = s_u [s * NH_ + d];
      float zs = s_z [s * NH_ + d];
      float du = dt * us;
      const float* Bp = &s_bc[s * (2 * NS_) + g * 32];
      const float* Cp = &s_bc[s * (2 * NS_) + NS_ + g * 32];
      float part = 0.0f;
#pragma unroll
      for (int i = 0; i < 32; i++) {
        float a  = __expf(dt * Arow[i]);
        float hn = a * h[i] + du * Bp[i];
        h[i] = hn;
        part += hn * Cp[i];
      }
      part += __shfl_xor(part, 1, 32);
      part += __shfl_xor(part, 2, 32);
      if (g == 0) {
        float y = (part + us * Dd) * siluf_(zs);
        y1_s[(size_t)(l0 + s) * NH_ + d] = y;
      }
    }
    __syncthreads();
  }
}

// ---------------- Kernel C: y@W_op -> SiLU -> @W_o + b_o ---------------------
__global__ __launch_bounds__(256) void k_outproj(
    const float* __restrict__ y1_g, const float* __restrict__ Wop,
    const float* __restrict__ Wo,  const float* __restrict__ bo,
    float* __restrict__ y2_g) {
  __shared__ __align__(16) float s_y[32 * NH_];
  __shared__ __align__(16) float s_t[32 * NH_];
  const int seq = blockIdx.x >> 4;
  const int l0  = (blockIdx.x & 15) * 32;
  const int tid = threadIdx.x;
  const int lane = tid & 31, w = tid >> 5;
  const int m = lane & 15, hh = lane >> 4;

  // async-stage the 32x64 y1 tile (8 KB, two B128 per thread)
  const float* src = y1_g + (size_t)seq * L_ * NH_ + (size_t)l0 * NH_;
  async_copy_b128(&s_y[tid * 8],     src + tid * 8);
  async_copy_b128(&s_y[tid * 8 + 4], src + tid * 8 + 4);
  async_wait0();
  __syncthreads();
  {
    int mi = w >> 2, ni = w & 3;
    v8f c = {};
    c = wmma_tile(&s_y[mi * 16 * NH_], NH_, Wop + ni * 16, NH_, NH_, c);
#pragma unroll
    for (int i = 0; i < 8; i++)
      s_t[(mi * 16 + i + 8 * hh) * NH_ + ni * 16 + m] = siluf_(c[i]);
  }
  __syncthreads();
  {
    int mi = w >> 2, ni = w & 3;
    v8f c = {};
    c = wmma_tile(&s_t[mi * 16 * NH_], NH_, Wo + ni * 16, NO_, NH_, c);
    float bb = bo[ni * 16 + m];
#pragma unroll
    for (int i = 0; i < 8; i++) {
      int row = l0 + mi * 16 + i + 8 * hh;
      y2_g[(size_t)seq * L_ * NO_ + (size_t)row * NO_ + ni * 16 + m] = c[i] + bb;
    }
  }
}

// ---------------- Kernel D: LayerNorm over (L,NOUT) + transposed write -------
__global__ __launch_bounds__(256) void k_layernorm(
    const float* __restrict__ y2_g, const float* __restrict__ lng,
    const float* __restrict__ lnb, float* __restrict__ out) {
  __shared__ float s_s[256], s_q[256];
  __shared__ float s_mu, s_inv;
  const int seq = blockIdx.x;
  const int tid = threadIdx.x;
  const int b = seq >> 4, node = seq & 15;
  const float* ys = y2_g + (size_t)seq * L_ * NO_;

  float sum = 0.0f, sq = 0.0f;
  for (int e = tid; e < L_ * NO_; e += 256) {
    float v = ys[e];
    sum += v; sq += v * v;
  }
  s_s[tid] = sum; s_q[tid] = sq;
  __syncthreads();
  for (int off = 128; off > 0; off >>= 1) {
    if (tid < off) { s_s[tid] += s_s[tid + off]; s_q[tid] += s_q[tid + off]; }
    __syncthreads();
  }
  if (tid == 0) {
    float mu  = s_s[0] * (1.0f / (L_ * NO_));
    float var = s_q[0] * (1.0f / (L_ * NO_)) - mu * mu;
    s_mu = mu;
    s_inv = rsqrtf(var + 1e-5f);
  }
  __syncthreads();
  const float mu = s_mu, inv = s_inv;
  // iterate (j, l) so consecutive threads write consecutive l (coalesced out)
  for (int e = tid; e < L_ * NO_; e += 256) {
    int j = e >> 9, l = e & 511;
    int li = l * NO_ + j;
    float v = (ys[li] - mu) * inv * lng[li] + lnb[li];
    out[(((size_t)node * NO_ + j) * 4 + b) * L_ + l] = v;
  }
}

extern "C" void kernel_launch(void* const* d_in, const int* in_sizes, int n_in,
                              void* d_out, int out_size, void* d_ws, size_t ws_size,
                              hipStream_t stream) {
  (void)in_sizes; (void)n_in; (void)out_size; (void)ws_size;
  const float* gin   = (const float*)d_in[0];
  const float* Wdim  = (const float*)d_in[1];
  const float* bdim  = (const float*)d_in[2];
  const float* Win   = (const float*)d_in[3];
  const float* convw = (const float*)d_in[4];
  const float* convb = (const float*)d_in[5];
  const float* Wxp   = (const float*)d_in[6];
  const float* Wdt   = (const float*)d_in[7];
  const float* bdt   = (const float*)d_in[8];
  const float* Alog  = (const float*)d_in[9];
  const float* Dskip = (const float*)d_in[10];
  const float* Wop   = (const float*)d_in[11];
  const float* Wo    = (const float*)d_in[12];
  const float* bo    = (const float*)d_in[13];
  const float* lng   = (const float*)d_in[14];
  const float* lnb   = (const float*)d_in[15];

  float* ws = (float*)d_ws;
  const size_t SEQELT = (size_t)L_ * NH_;   // 32768 floats per sequence
  float* xi = ws;
  float* z  = ws + 64 * SEQELT;
  float* y1 = ws + 2 * 64 * SEQELT;
  float* y2 = ws + 3 * 64 * SEQELT;        // total 8 Mi floats = 32 MB

  k_inproj   <<<dim3(1024), dim3(256), 0, stream>>>(gin, Wdim, bdim, Win, xi, z);
  k_scan     <<<dim3(64),   dim3(256), 0, stream>>>(xi, z, convw, convb, Wxp, Wdt,
                                                    bdt, Alog, Dskip, y1);
  k_outproj  <<<dim3(1024), dim3(256), 0, stream>>>(y1, Wop, Wo, bo, y2);
  k_layernorm<<<dim3(64),   dim3(256), 0, stream>>>(y2, lng, lnb, (float*)d_out);
}